// GPRGNN_15530601743023
// MI455X (gfx1250) — compile-verified
//
#include <hip/hip_runtime.h>
#include <hip/hip_bf16.h>
#include <math.h>

// ---------------- problem constants (match reference) ----------------
#define N_NODES   100000
#define NFEAT     512
#define HID       64
#define NCLS      40
#define N_EDGES   3200000
#define KSTEPS    10

typedef float v2f __attribute__((ext_vector_type(2)));
typedef float v8f __attribute__((ext_vector_type(8)));

#define XS_STRIDE 516   // 516 % 64 == 4 -> 16 rows map to 16 distinct LDS banks
#define HS_STRIDE 68    // 68  % 64 == 4 -> same trick for the hidden tile

// ===================== degree / norm prep =====================
__global__ void k_deg_init(float* __restrict__ deg) {
    int i = blockIdx.x * blockDim.x + threadIdx.x;
    if (i < N_NODES) deg[i] = 1.0f;  // self-loop weight
}

__global__ void k_deg_edges(const int* __restrict__ dst, float* __restrict__ deg) {
    int e = blockIdx.x * blockDim.x + threadIdx.x;
    if (e < N_EDGES) atomicAdd(&deg[dst[e]], 1.0f);
}

__global__ void k_dinv(float* __restrict__ deg) {
    int i = blockIdx.x * blockDim.x + threadIdx.x;
    if (i < N_NODES) {
        float d = deg[i];
        deg[i] = (d > 0.0f) ? rsqrtf(d) : 0.0f;  // in-place deg -> dinv
    }
}

__global__ void k_norm(const int* __restrict__ src, const int* __restrict__ dst,
                       const float* __restrict__ dinv, float* __restrict__ norm) {
    int e = blockIdx.x * blockDim.x + threadIdx.x;
    if (e < N_EDGES) norm[e] = dinv[src[e]] * dinv[dst[e]];
}

// ===================== fused MLP via V_WMMA_F32_16X16X4_F32 =====================
// One 128-thread (4 wave32) block per 16 node rows.
// GEMM1: [16 x 512] x [512 x 64], each wave owns a 16-wide hidden column tile.
// GEMM2: [16 x 64] x [64 x 48(pad)] on waves 0..2, valid cols < 40.
__launch_bounds__(128)
__global__ void k_mlp(const float* __restrict__ x,  const float* __restrict__ w1,
                      const float* __restrict__ b1, const float* __restrict__ w2,
                      const float* __restrict__ b2, float* __restrict__ h) {
    __shared__ float xs[16 * XS_STRIDE];
    __shared__ float hs[16 * HS_STRIDE];

    const int tid  = threadIdx.x;
    const int wave = tid >> 5;
    const int lane = tid & 31;
    const int row0 = blockIdx.x * 16;

    // stage 16x512 x-tile (contiguous 32KB) with fully coalesced loads
    const float* xg = x + (size_t)row0 * NFEAT;
    for (int i = tid; i < 16 * NFEAT; i += 128) {
        int r = i >> 9;          // /512
        int c = i & (NFEAT - 1); // %512
        xs[r * XS_STRIDE + c] = xg[i];
    }
    __syncthreads();

    // ISA A-frag layout (32-bit 16x4): lanes 0-15 -> K+{0,1}, lanes 16-31 -> K+{2,3}
    const int n    = lane & 15;        // column within tile / row M for A
    const int kofs = (lane >> 4) * 2;  // K sub-offset
    const int rsel = (lane >> 4) * 8;  // C/D row group per ISA 16x16 f32 layout
    const int nc0  = wave * 16;

    // ---- GEMM1: h1 = relu(x @ w1 + b1) ----
    v8f acc = {0.f, 0.f, 0.f, 0.f, 0.f, 0.f, 0.f, 0.f};
    for (int k = 0; k < NFEAT; k += 4) {
        v2f a, b;
        a.x = xs[n * XS_STRIDE + k + kofs];
        a.y = xs[n * XS_STRIDE + k + kofs + 1];
        b.x = w1[(k + kofs)     * HID + nc0 + n];
        b.y = w1[(k + kofs + 1) * HID + nc0 + n];
        acc = __builtin_amdgcn_wmma_f32_16x16x4_f32(false, a, false, b,
                                                    (short)0, acc, false, false);
    }
    const float bias1 = b1[nc0 + n];
#pragma unroll
    for (int v = 0; v < 8; ++v) {
        float hv = acc[v] + bias1;
        hv = hv > 0.0f ? hv : 0.0f;
        hs[(v + rsel) * HS_STRIDE + nc0 + n] = hv;
    }
    __syncthreads();

    // ---- GEMM2: out = h1 @ w2 + b2 (40 cols padded to 48) ----
    if (wave < 3) {
        v8f acc2 = {0.f, 0.f, 0.f, 0.f, 0.f, 0.f, 0.f, 0.f};
        const int col = nc0 + n;
        const bool ok = (col < NCLS);
        for (int k = 0; k < HID; k += 4) {
            v2f a, b;
            a.x = hs[n * HS_STRIDE + k + kofs];
            a.y = hs[n * HS_STRIDE + k + kofs + 1];
            b.x = ok ? w2[(k + kofs)     * NCLS + col] : 0.0f;
            b.y = ok ? w2[(k + kofs + 1) * NCLS + col] : 0.0f;
            acc2 = __builtin_amdgcn_wmma_f32_16x16x4_f32(false, a, false, b,
                                                         (short)0, acc2, false, false);
        }
        if (ok) {
            const float bias2 = b2[col];
#pragma unroll
            for (int v = 0; v < 8; ++v) {
                int r = row0 + v + rsel;
                h[(size_t)r * NCLS + col] = acc2[v] + bias2;
            }
        }
    }
}

// ===================== propagation =====================
// out = gamma0 * h   (float4 over N*40)
__global__ void k_out_init(const float4* __restrict__ h4, float4* __restrict__ out4,
                           const float* __restrict__ temp) {
    int t = blockIdx.x * blockDim.x + threadIdx.x;
    if (t >= N_NODES * (NCLS / 4)) return;
    float g = temp[0];
    g = g > 0.0f ? g : 0.0f;
    float4 v = h4[t];
    out4[t] = make_float4(g * v.x, g * v.y, g * v.z, g * v.w);
}

// edge scatter: nxt[dst] += norm_e * cur[src]; 10 threads per edge (float4 each)
__global__ void k_scatter(const int* __restrict__ src, const int* __restrict__ dst,
                          const float* __restrict__ norm,
                          const float* __restrict__ cur, float* __restrict__ nxt) {
    int t = blockIdx.x * blockDim.x + threadIdx.x;
    if (t >= N_EDGES * 10) return;
    int e = t / 10;
    int q = t - e * 10;
    int s = src[e];
    int d = dst[e];
    float w = norm[e];
    float4 v = ((const float4*)cur)[s * 10 + q];
    float* o = nxt + (size_t)d * NCLS + q * 4;
    atomicAdd(o + 0, w * v.x);
    atomicAdd(o + 1, w * v.y);
    atomicAdd(o + 2, w * v.z);
    atomicAdd(o + 3, w * v.w);
}

// nxt += dinv[i]^2 * cur (self loop); out += gamma_k * nxt
__global__ void k_self_acc(const float* __restrict__ dinv,
                           const float4* __restrict__ cur4, float4* __restrict__ nxt4,
                           float4* __restrict__ out4, const float* __restrict__ temp,
                           int kidx) {
    int t = blockIdx.x * blockDim.x + threadIdx.x;
    if (t >= N_NODES * (NCLS / 4)) return;
    int i = t / 10;
    float di = dinv[i];
    float sw = di * di;
    float4 c = cur4[t];
    float4 v = nxt4[t];
    v.x += sw * c.x; v.y += sw * c.y; v.z += sw * c.z; v.w += sw * c.w;
    nxt4[t] = v;
    float g = temp[kidx];
    g = g > 0.0f ? g : 0.0f;
    float4 o = out4[t];
    o.x += g * v.x; o.y += g * v.y; o.z += g * v.z; o.w += g * v.w;
    out4[t] = o;
}

// ===================== log-softmax, one wave32 per node =====================
__global__ void k_logsoftmax(float* __restrict__ out) {
    int wave = threadIdx.x >> 5;
    int lane = threadIdx.x & 31;
    int i = blockIdx.x * 8 + wave;
    if (i >= N_NODES) return;
    float* row = out + (size_t)i * NCLS;
    float v0 = row[lane];
    float v1 = (lane < 8) ? row[32 + lane] : -3.402823466e38f;
    float mx = fmaxf(v0, v1);
    for (int o = 16; o > 0; o >>= 1) mx = fmaxf(mx, __shfl_xor(mx, o, 32));
    float s = __expf(v0 - mx) + ((lane < 8) ? __expf(v1 - mx) : 0.0f);
    for (int o = 16; o > 0; o >>= 1) s += __shfl_xor(s, o, 32);
    float lse = mx + __logf(s);
    row[lane] = v0 - lse;
    if (lane < 8) row[32 + lane] = v1 - lse;
}

// ===================== host orchestration =====================
extern "C" void kernel_launch(void* const* d_in, const int* in_sizes, int n_in,
                              void* d_out, int out_size, void* d_ws, size_t ws_size,
                              hipStream_t stream) {
    const float* x    = (const float*)d_in[0];
    const int*   ei   = (const int*)d_in[1];   // [2, E] int32
    const float* w1   = (const float*)d_in[2];
    const float* b1   = (const float*)d_in[3];
    const float* w2   = (const float*)d_in[4];
    const float* b2   = (const float*)d_in[5];
    const float* temp = (const float*)d_in[6];
    const int* src = ei;
    const int* dst = ei + N_EDGES;

    float* out = (float*)d_out;  // N*40

    // workspace layout (256B-aligned chunks)
    char* ws = (char*)d_ws;
    const size_t featBytes = (size_t)N_NODES * NCLS * sizeof(float);   // 16 MB
    float* bufA = (float*)(ws);                         // h / ping
    float* bufB = (float*)(ws + featBytes);             // pong
    float* dinv = (float*)(ws + 2 * featBytes);         // deg -> dinv, N floats
    float* norm = (float*)(ws + 2 * featBytes + ((size_t)N_NODES * 4 + 255) / 256 * 256);

    const int TB = 256;

    // 1) degrees + dinv + per-edge norms
    k_deg_init <<<(N_NODES + TB - 1) / TB, TB, 0, stream>>>(dinv);
    k_deg_edges<<<(N_EDGES + TB - 1) / TB, TB, 0, stream>>>(dst, dinv);
    k_dinv     <<<(N_NODES + TB - 1) / TB, TB, 0, stream>>>(dinv);
    k_norm     <<<(N_EDGES + TB - 1) / TB, TB, 0, stream>>>(src, dst, dinv, norm);

    // 2) fused MLP (WMMA), h -> bufA
    k_mlp<<<N_NODES / 16, 128, 0, stream>>>(x, w1, b1, w2, b2, bufA);

    // 3) out = relu(temp[0]) * h
    const int nv4 = N_NODES * (NCLS / 4);
    k_out_init<<<(nv4 + TB - 1) / TB, TB, 0, stream>>>((const float4*)bufA,
                                                       (float4*)out, temp);

    // 4) K propagation steps (ping-pong bufA/bufB)
    float* cur = bufA;
    float* nxt = bufB;
    for (int k = 0; k < KSTEPS; ++k) {
        hipMemsetAsync(nxt, 0, featBytes, stream);
        k_scatter<<<(N_EDGES * 10 + TB - 1) / TB, TB, 0, stream>>>(src, dst, norm, cur, nxt);
        k_self_acc<<<(nv4 + TB - 1) / TB, TB, 0, stream>>>(dinv, (const float4*)cur,
                                                           (float4*)nxt, (float4*)out,
                                                           temp, k + 1);
        float* t2 = cur; cur = nxt; nxt = t2;
    }

    // 5) log-softmax in place on d_out
    k_logsoftmax<<<(N_NODES + 7) / 8, 256, 0, stream>>>(out);
}